// KIWDBallChanger_53017076302315
// MI455X (gfx1250) — compile-verified
//
#include <hip/hip_runtime.h>
#include <stdint.h>

// ---------------------------------------------------------------------------
// IDW interpolation over a COO radius graph (MI455X / gfx1250).
//
// Roofline: 51.2 MB of edge indices streamed once (~2.2us @ 23.3 TB/s);
// positions (2.4MB), x (3.2MB) and accumulators (3.6MB) are L2-resident
// (192MB L2). Limiter is 57.6M f32 atomics into L2 -> minimize gather cost
// per edge and stream the edge lists via the CDNA5 async global->LDS path
// (ASYNCcnt), double-buffered as a decoupled prefetch.
// No matmul structure exists (K=3, random scatter) -> WMMA not applicable.
// ---------------------------------------------------------------------------

#define TPB      256          // threads per block (8 wave32)
#define VEC      4            // edges per thread per chunk (one b128 per list)
#define CHUNK_E  (TPB * VEC)  // 1024 edges per chunk
#define NCHUNK   8            // chunks per block
#define EPB      (CHUNK_E * NCHUNK)  // 8192 edges per block

typedef int v4i __attribute__((ext_vector_type(4)));
typedef __attribute__((address_space(1))) v4i* as1_v4i_p;
typedef __attribute__((address_space(3))) v4i* as3_v4i_p;

// 16B async copy global -> LDS (GLOBAL_LOAD_ASYNC_TO_LDS_B128, ASYNCcnt)
__device__ __forceinline__ void async_b128(const void* g, void* l) {
#if __has_builtin(__builtin_amdgcn_global_load_async_to_lds_b128)
  // generic global address == AS1 address; low 32 bits of a generic LDS
  // pointer are the LDS byte offset (AS3 address)
  __builtin_amdgcn_global_load_async_to_lds_b128(
      (as1_v4i_p)(uintptr_t)g, (as3_v4i_p)(uint32_t)(uintptr_t)l, 0, 0);
#else
  asm volatile("global_load_async_to_lds_b128 %0, %1, off"
               :: "v"((uint32_t)(uintptr_t)l), "v"(g)
               : "memory");
#endif
}

template <int N>
__device__ __forceinline__ void wait_async() {
#if __has_builtin(__builtin_amdgcn_s_wait_asynccnt)
  __builtin_amdgcn_s_wait_asynccnt(N);
#else
  asm volatile("s_wait_asynccnt %0" :: "i"(N) : "memory");
#endif
  // Compiler-level memory barrier: the async engine wrote LDS behind the
  // compiler's back (builtin pointer came from an int cast), so force all
  // later LDS reads to actually load (prevents undef-folding of s_in/s_out).
  asm volatile("" ::: "memory");
}

// ---------------------------------------------------------------------------

__global__ __launch_bounds__(TPB) void edge_scatter_kernel(
    const int* __restrict__ ein, const int* __restrict__ eout,
    const float* __restrict__ pin, const float* __restrict__ pout,
    const float4* __restrict__ xt, const float* __restrict__ x,
    float* __restrict__ num, float* __restrict__ dem,
    int E, int n_in, int n_out) {
  __shared__ int s_in[2][CHUNK_E];
  __shared__ int s_out[2][CHUNK_E];

  const int tid  = threadIdx.x;
  const int base = blockIdx.x * EPB;
  const int mx   = E - VEC;  // clamp point so b128 stays in-bounds

  auto issue = [&](int c) {
    const int buf = c & 1;
    int idx = base + c * CHUNK_E + tid * VEC;
    if (idx > mx) idx = mx;
    async_b128(ein + idx,  &s_in[buf][tid * VEC]);
    async_b128(eout + idx, &s_out[buf][tid * VEC]);
  };

  issue(0);
  for (int c = 0; c < NCHUNK; ++c) {
    const int buf = c & 1;
    if (c + 1 < NCHUNK) {
      issue(c + 1);     // prefetch next chunk (lane-private slots, no barrier)
      wait_async<2>();  // oldest 2 async ops (this chunk) complete, in-order
    } else {
      wait_async<0>();
    }

    // One ds_load_b128 per list per chunk (16B-aligned slices).
    const int4 vsrc = *reinterpret_cast<const int4*>(&s_in[buf][tid * VEC]);
    const int4 vdst = *reinterpret_cast<const int4*>(&s_out[buf][tid * VEC]);
    int srcs[VEC] = {vsrc.x, vsrc.y, vsrc.z, vsrc.w};
    int dsts[VEC] = {vdst.x, vdst.y, vdst.z, vdst.w};

    const int estart   = base + c * CHUNK_E + tid * VEC;
    const bool shifted = (estart > mx);  // rare straddling-tail thread

#pragma unroll
    for (int k = 0; k < VEC; ++k) {
      const int e = estart + k;
      if (e >= E) break;
      int src = srcs[k];
      int dst = dsts[k];
      if (shifted) {                 // clamped copy doesn't match this edge
        src = ein[e];
        dst = eout[e];
      }

      const int s3 = src * 3, d3 = dst * 3;
      const float dx = 1.0f + pin[s3 + 0] - pout[d3 + 0];
      const float dy = 1.0f + pin[s3 + 1] - pout[d3 + 1];
      const float dz = 1.0f + pin[s3 + 2] - pout[d3 + 2];
      const float w  = 1.0f / sqrtf(dx * dx + dy * dy + dz * dz);

      atomicAdd(dem + dst, w);
      if (xt) {
        const float4 v0 = xt[src * 2 + 0];
        const float4 v1 = xt[src * 2 + 1];
        atomicAdd(num + 0 * n_out + dst, w * v0.x);
        atomicAdd(num + 1 * n_out + dst, w * v0.y);
        atomicAdd(num + 2 * n_out + dst, w * v0.z);
        atomicAdd(num + 3 * n_out + dst, w * v0.w);
        atomicAdd(num + 4 * n_out + dst, w * v1.x);
        atomicAdd(num + 5 * n_out + dst, w * v1.y);
        atomicAdd(num + 6 * n_out + dst, w * v1.z);
        atomicAdd(num + 7 * n_out + dst, w * v1.w);
      } else {
#pragma unroll
        for (int b = 0; b < 8; ++b)
          atomicAdd(num + b * n_out + dst, w * x[b * n_in + src]);
      }
    }
  }
}

__global__ __launch_bounds__(TPB) void zero_f32_kernel(float* p, int n) {
  int i = blockIdx.x * TPB + threadIdx.x;
  if (i < n) p[i] = 0.0f;
}

// x[8][n_in] -> xt[n_in][8] so the 8 per-source field values live in 32
// contiguous bytes (two b128 loads per edge instead of 8 scattered loads).
__global__ __launch_bounds__(TPB) void transpose_x_kernel(
    const float* __restrict__ x, float4* __restrict__ xt, int n_in) {
  int i = blockIdx.x * TPB + threadIdx.x;
  if (i >= n_in) return;
  float4 a, b;
  a.x = x[0 * n_in + i]; a.y = x[1 * n_in + i];
  a.z = x[2 * n_in + i]; a.w = x[3 * n_in + i];
  b.x = x[4 * n_in + i]; b.y = x[5 * n_in + i];
  b.z = x[6 * n_in + i]; b.w = x[7 * n_in + i];
  xt[i * 2 + 0] = a;
  xt[i * 2 + 1] = b;
}

__global__ __launch_bounds__(TPB) void normalize_kernel(
    float* __restrict__ out, const float* __restrict__ dem, int n_out) {
  int t = blockIdx.x * TPB + threadIdx.x;
  if (t >= n_out) return;
  float d = dem[t];
  d = (d > 0.0f) ? d : 1.0f;
#pragma unroll
  for (int b = 0; b < 8; ++b)
    out[b * n_out + t] = out[b * n_out + t] / d;
}

// ---------------------------------------------------------------------------

extern "C" void kernel_launch(void* const* d_in, const int* in_sizes, int n_in_args,
                              void* d_out, int out_size, void* d_ws, size_t ws_size,
                              hipStream_t stream) {
  (void)n_in_args; (void)out_size;
  const float* x    = (const float*)d_in[0];   // [8, n_in]
  const float* pin  = (const float*)d_in[1];   // [n_in, 3]
  const float* pout = (const float*)d_in[2];   // [n_out, 3]
  const int*   eout = (const int*)d_in[3];     // [E] targets
  const int*   ein  = (const int*)d_in[4];     // [E] sources

  const int n_in  = in_sizes[1] / 3;
  const int n_out = in_sizes[2] / 3;
  const int E     = in_sizes[3];

  float* out = (float*)d_out;                  // [8, n_out] accumulates num
  float* dem = (float*)d_ws;                   // [n_out]

  const size_t dem_bytes = (((size_t)n_out * sizeof(float)) + 255) & ~(size_t)255;
  const size_t xt_bytes  = (size_t)n_in * 8 * sizeof(float);
  const bool   use_xt    = ws_size >= dem_bytes + xt_bytes;
  float4* xt = use_xt ? (float4*)((char*)d_ws + dem_bytes) : nullptr;

  zero_f32_kernel<<<(8 * n_out + TPB - 1) / TPB, TPB, 0, stream>>>(out, 8 * n_out);
  zero_f32_kernel<<<(n_out + TPB - 1) / TPB, TPB, 0, stream>>>(dem, n_out);
  if (use_xt)
    transpose_x_kernel<<<(n_in + TPB - 1) / TPB, TPB, 0, stream>>>(x, xt, n_in);

  const int nblk = (E + EPB - 1) / EPB;
  edge_scatter_kernel<<<nblk, TPB, 0, stream>>>(ein, eout, pin, pout, xt, x,
                                                out, dem, E, n_in, n_out);

  normalize_kernel<<<(n_out + TPB - 1) / TPB, TPB, 0, stream>>>(out, dem, n_out);
}